// QuantumInspiredCryptanalysis_1726576853344
// MI455X (gfx1250) — compile-verified
//
#include <hip/hip_runtime.h>
#include <hip/hip_bf16.h>
#include <math.h>

typedef __attribute__((ext_vector_type(2))) float v2f;
typedef __attribute__((ext_vector_type(8))) float v8f;

#define THREADS 64
#define TOKENS_PER_BLOCK 64

__device__ __forceinline__ float gelu_erf(float v) {
    // exact-erf GELU, matches jax.nn.gelu(approximate=False)
    return 0.5f * v * (1.0f + erff(v * 0.70710678118654752f));
}

__global__ __launch_bounds__(THREADS)
void qic_fused(const float* __restrict__ x,        // [B*S] = 65536
               const float* __restrict__ enc_w1,   // [8]
               const float* __restrict__ enc_b1,   // [8]
               const float* __restrict__ enc_w2,   // [8,256]
               const float* __restrict__ enc_b2,   // [256]
               const float* __restrict__ rot,      // [8,3]
               const float* __restrict__ ent,      // [7]
               const float* __restrict__ gate,     // [3,8,4]
               const float* __restrict__ M_re,     // [256,256] (only [0][0] live)
               const float* __restrict__ cw1,      // [256,256] (only row 0 live)
               const float* __restrict__ cb1,      // [256]
               const float* __restrict__ cw2,      // [256,128]
               const float* __restrict__ cb2,      // [128]
               const float* __restrict__ cw3,      // [128]
               const float* __restrict__ cb3,      // [1]
               float* __restrict__ out)            // [65536]
{
    __shared__ float s_cw0sq[TOKENS_PER_BLOCK];
    __shared__ float s_scale;
    // 32 KB buffer: cw2 K-chunk during the matmul, then reused for h2pre
    __shared__ __align__(16) float s_b[64 * 128];

    const int tid    = threadIdx.x;
    const int wgbase = blockIdx.x * TOKENS_PER_BLOCK;

    // ---- Phase 0: |enc_scal * proc_scal|^2 * M_re[0,0] (one thread; ~150 flops)
    if (tid == 0) {
        float z2 = M_re[0];
        #pragma unroll 1
        for (int q = 0; q < 8; ++q) {
            float h0 = 0.5f * rot[q*3+0], h1 = 0.5f * rot[q*3+1], h2 = 0.5f * rot[q*3+2];
            float re = cosf(h0) * cosf(h1) * cosf(h2);
            float im = sinf(h0) * sinf(h1) * sinf(h2);
            z2 *= re*re + im*im;
        }
        #pragma unroll 1
        for (int j = 0; j < 7; ++j) {
            float g = 1.0f / (1.0f + expf(-ent[j]));
            z2 *= g*g + (1.0f - g)*(1.0f - g);
        }
        #pragma unroll 1
        for (int t = 0; t < 24; ++t) {
            float a  = gate[t*4+0], b = gate[t*4+1], gm = gate[t*4+2];
            float re = 2.0f * cosf(a) * cosf(b);
            float im = 2.0f * sinf(a) * sinf(gm);
            z2 *= re*re + im*im;
        }
        s_scale = z2;
    }

    // ---- Phase 1: per-token softmax weight cw0^2 (wave-uniform w2/b2 -> s_loads)
    {
        const int tok = wgbase + tid;
        const float xv = x[tok];
        float hq[8];
        #pragma unroll
        for (int q = 0; q < 8; ++q)
            hq[q] = tanhf(fmaf(xv, enc_w1[q], enc_b1[q]));

        // k = 0 logit, peeled
        float l0 = enc_b2[0];
        #pragma unroll
        for (int q = 0; q < 8; ++q) l0 = fmaf(hq[q], enc_w2[q*256], l0);

        float mx = l0;
        #pragma clang loop unroll_count(8)
        for (int k = 1; k < 256; ++k) {
            float l = enc_b2[k];
            #pragma unroll
            for (int q = 0; q < 8; ++q) l = fmaf(hq[q], enc_w2[q*256 + k], l);
            mx = fmaxf(mx, l);
        }
        float sum = 0.0f;
        #pragma clang loop unroll_count(8)
        for (int k = 0; k < 256; ++k) {
            float l = enc_b2[k];
            #pragma unroll
            for (int q = 0; q < 8; ++q) l = fmaf(hq[q], enc_w2[q*256 + k], l);
            sum += expf(l - mx);
        }
        const float cw0 = expf(l0 - mx) / sum;
        s_cw0sq[tid] = cw0 * cw0;
    }
    __syncthreads();

    // ---- Phase 2: h2pre = h1[64,256] @ cw2[256,128] via V_WMMA_F32_16X16X4_F32
    const int lane = tid & 31;
    const int wave = tid >> 5;
    const int n    = lane & 15;     // N col within 16x16 tile / A row
    const int hi   = lane >> 4;     // lane half
    const int kb   = hi << 1;       // K-pair select: {0,1} vs {2,3}

    const float scale = s_scale;
    const float m0 = scale * s_cw0sq[wave*32 + n];        // token row, M-tile 0
    const float m1 = scale * s_cw0sq[wave*32 + 16 + n];   // token row, M-tile 1

    v8f acc[2][8] = {};   // [mtile][ntile] 16x16 f32 C/D fragments

    #pragma unroll 1
    for (int k0 = 0; k0 < 256; k0 += 64) {
        __syncthreads();
        // stage cw2 rows [k0, k0+64) -> LDS; one 128-float row per thread
        {
            const float4* src = (const float4*)(cw2 + (k0 + tid) * 128);
            float4*       dst = (float4*)(s_b + tid * 128);
            #pragma unroll 8
            for (int i = 0; i < 32; ++i) dst[i] = src[i];
        }
        __syncthreads();

        #pragma unroll 1
        for (int kk = 0; kk < 64; kk += 4) {
            const int kg = k0 + kk + kb;   // global K index for this lane half
            const int kl = kk + kb;        // LDS row
            // A fragment (h1 computed on the fly; VALU co-executes with WMMA)
            const float wa = cw1[kg], wb = cw1[kg + 1];
            const float ba = cb1[kg], bb = cb1[kg + 1];
            v2f a0, a1;
            a0.x = gelu_erf(fmaf(m0, wa, ba));
            a0.y = gelu_erf(fmaf(m0, wb, bb));
            a1.x = gelu_erf(fmaf(m1, wa, ba));
            a1.y = gelu_erf(fmaf(m1, wb, bb));
            #pragma unroll
            for (int nt = 0; nt < 8; ++nt) {
                v2f b;
                b.x = s_b[kl * 128 + nt*16 + n];
                b.y = s_b[(kl + 1) * 128 + nt*16 + n];
                acc[0][nt] = __builtin_amdgcn_wmma_f32_16x16x4_f32(
                    false, a0, false, b, (short)0, acc[0][nt], false, false);
                acc[1][nt] = __builtin_amdgcn_wmma_f32_16x16x4_f32(
                    false, a1, false, b, (short)0, acc[1][nt], false, false);
            }
        }
    }

    // ---- Phase 3a: dump C fragments to LDS (s_b reused as h2pre[64][128]).
    // C layout: col = lane%16, vgpr v <-> row v (lanes 0-15) / v+8 (lanes 16-31).
    __syncthreads();   // all waves done reading the last cw2 chunk
    #pragma unroll
    for (int mt = 0; mt < 2; ++mt) {
        const int rowbase = wave*32 + mt*16 + 8*hi;
        #pragma unroll
        for (int nt = 0; nt < 8; ++nt) {
            const int col = nt*16 + n;
            #pragma unroll
            for (int v = 0; v < 8; ++v)
                s_b[(rowbase + v)*128 + col] = acc[mt][nt][v];
        }
    }
    __syncthreads();

    // ---- Phase 3b: out[tok] = sum_i gelu(h2pre[i]+cb2[i]) * cw3[i] + cb3
    // One token per thread, rolled loop (keeps the 128 erf expansions out of I$),
    // left-to-right summation matches the reference's reduction order.
    {
        float s = 0.0f;
        const float* h2row = s_b + tid * 128;
        #pragma clang loop unroll_count(4)
        for (int i = 0; i < 128; ++i)
            s += gelu_erf(h2row[i] + cb2[i]) * cw3[i];
        out[wgbase + tid] = s + cb3[0];
    }
}

extern "C" void kernel_launch(void* const* d_in, const int* in_sizes, int n_in,
                              void* d_out, int out_size, void* d_ws, size_t ws_size,
                              hipStream_t stream) {
    const float* x      = (const float*)d_in[0];
    const float* enc_w1 = (const float*)d_in[1];
    const float* enc_b1 = (const float*)d_in[2];
    const float* enc_w2 = (const float*)d_in[3];
    const float* enc_b2 = (const float*)d_in[4];
    const float* rot    = (const float*)d_in[5];
    const float* ent    = (const float*)d_in[6];
    const float* gate   = (const float*)d_in[7];
    const float* M_re   = (const float*)d_in[8];
    // d_in[9] = M_im is provably dead (state is a delta at index 0; measurement
    // is elementwise, so only Re(M[0,0]) survives).
    const float* cw1    = (const float*)d_in[10];
    const float* cb1    = (const float*)d_in[11];
    const float* cw2    = (const float*)d_in[12];
    const float* cb2    = (const float*)d_in[13];
    const float* cw3    = (const float*)d_in[14];
    const float* cb3    = (const float*)d_in[15];
    float* out = (float*)d_out;

    const int tokens = in_sizes[0];                 // 32 * 2048 = 65536
    const int blocks = tokens / TOKENS_PER_BLOCK;   // 1024
    qic_fused<<<blocks, THREADS, 0, stream>>>(x, enc_w1, enc_b1, enc_w2, enc_b2,
                                              rot, ent, gate, M_re,
                                              cw1, cb1, cw2, cb2, cw3, cb3, out);
}